// HGPSLGNN_NotConnected_46033459478726
// MI455X (gfx1250) — compile-verified
//
#include <hip/hip_runtime.h>

// ---------------------------------------------------------------------------
// HGPSL GNN forward for MI455X (gfx1250, wave32).
//  * Node MLP GEMMs via V_WMMA_F32_16X16X4_F32 (full fp32 tensor ops).
//  * GCN aggregation via CSR-by-dst build (int atomics + scan) + pure gather:
//    no fp32 atomics anywhere; gather streams h[src] from L2 (h = 32MB << 192MB).
// ---------------------------------------------------------------------------

typedef float v2f __attribute__((ext_vector_type(2)));
typedef float v8f __attribute__((ext_vector_type(8)));

#define NNODES 131072
#define NEDGES 2097152
#define NGRAPH 128
#define NPG    1024     // nodes per graph
#define FD     64       // feature dim everywhere in the branch
#define AST    68       // LDS activation row stride (bank-conflict-free)
#define CHUNKS (NNODES / 256)   // 512 scan chunks

// ---- one 16x64 = 16x64 @ 64x64 layer slice via WMMA f32 16x16x4 ------------
__device__ __forceinline__ void wmma_layer16x64(const float* S, int m0,
                                                const float* __restrict__ W,
                                                v8f acc[4]) {
  const int lane = threadIdx.x & 31;
  const int m    = lane & 15;          // A row within tile == B column within tile
  const int kl   = (lane >> 4) << 1;   // hi-half lanes take K pair {2,3}
  const float* Srow = S + (m0 + m) * AST;
#pragma unroll
  for (int nt = 0; nt < 4; ++nt) {
    v8f c = {};
    const int n = (nt << 4) + m;
#pragma unroll
    for (int ks = 0; ks < 16; ++ks) {
      const int k0 = (ks << 2) + kl;
      v2f a, b;
      a.x = Srow[k0];
      a.y = Srow[k0 + 1];
      b.x = W[k0 * FD + n];
      b.y = W[(k0 + 1) * FD + n];
      c = __builtin_amdgcn_wmma_f32_16x16x4_f32(false, a, false, b,
                                                (short)0, c, false, false);
    }
    acc[nt] = c;
  }
}

__device__ __forceinline__ void store_act_relu(float* O, int m0, const v8f acc[4],
                                               const float* __restrict__ bias) {
  const int lane = threadIdx.x & 31;
  const int l15  = lane & 15;
  const int hi   = lane >> 4;
#pragma unroll
  for (int nt = 0; nt < 4; ++nt) {
    const int n  = (nt << 4) + l15;
    const float bv = bias[n];
#pragma unroll
    for (int r = 0; r < 8; ++r) {
      const int mr = m0 + (hi << 3) + r;       // ISA C layout: row = r + 8*hi
      O[mr * AST + n] = fmaxf(acc[nt][r] + bv, 0.0f);
    }
  }
}

// h = relu(relu(x@W1+b1)@W2+b2) @ Wg     (GCN bias added after aggregation)
__global__ void __launch_bounds__(128)
mlp_branch(const float* __restrict__ x,
           const float* __restrict__ W1, const float* __restrict__ b1,
           const float* __restrict__ W2, const float* __restrict__ b2,
           const float* __restrict__ Wg,
           float* __restrict__ h_out) {
  __shared__ float actA[FD * AST];
  __shared__ float actB[FD * AST];
  const int tid  = threadIdx.x;
  const int lane = tid & 31;
  const int wave = tid >> 5;
  const long rowBase = (long)blockIdx.x * 64;

  for (int i = tid; i < FD * FD; i += 128) {
    const int r = i >> 6, c = i & 63;
    actA[r * AST + c] = x[(rowBase + r) * FD + c];
  }
  __syncthreads();

  const int m0 = wave << 4;   // each wave owns 16 rows -> fully wave-local layers
  v8f acc[4];

  wmma_layer16x64(actA, m0, W1, acc);
  store_act_relu(actB, m0, acc, b1);

  wmma_layer16x64(actB, m0, W2, acc);
  store_act_relu(actA, m0, acc, b2);

  wmma_layer16x64(actA, m0, Wg, acc);

  const int l15 = lane & 15;
  const int hi  = lane >> 4;
#pragma unroll
  for (int nt = 0; nt < 4; ++nt) {
    const int n = (nt << 4) + l15;
#pragma unroll
    for (int r = 0; r < 8; ++r) {
      const int mr = m0 + (hi << 3) + r;
      h_out[(rowBase + mr) * FD + n] = acc[nt][r];
    }
  }
}

// ---- CSR-by-destination build ---------------------------------------------
__global__ void zero_ints(int* __restrict__ p, int n) {
  int i = blockIdx.x * blockDim.x + threadIdx.x;
  if (i < n) p[i] = 0;
}

__global__ void cnt_scatter(const int* __restrict__ dst, int* __restrict__ cnt,
                            int ne) {
  int e = blockIdx.x * blockDim.x + threadIdx.x;
  if (e < ne) atomicAdd(&cnt[dst[e]], 1);
}

__global__ void __launch_bounds__(256)
chunk_reduce(const int* __restrict__ cnt, int* __restrict__ chunkSum) {
  __shared__ int s[256];
  const int b = blockIdx.x, t = threadIdx.x;
  s[t] = cnt[b * 256 + t];
  __syncthreads();
  for (int d = 128; d > 0; d >>= 1) {
    if (t < d) s[t] += s[t + d];
    __syncthreads();
  }
  if (t == 0) chunkSum[b] = s[0];
}

__global__ void __launch_bounds__(512)
chunk_scan(const int* __restrict__ chunkSum, int* __restrict__ chunkOff) {
  __shared__ int s[CHUNKS];
  const int t = threadIdx.x;
  const int v = chunkSum[t];
  s[t] = v;
  __syncthreads();
  for (int d = 1; d < CHUNKS; d <<= 1) {       // Hillis-Steele inclusive
    int x = (t >= d) ? s[t - d] : 0;
    __syncthreads();
    s[t] += x;
    __syncthreads();
  }
  chunkOff[t] = s[t] - v;                      // exclusive
}

__global__ void __launch_bounds__(256)
node_scan(const int* __restrict__ cnt, const int* __restrict__ chunkOff,
          int* __restrict__ nodeOff) {
  __shared__ int s[256];
  const int b = blockIdx.x, t = threadIdx.x;
  const int v = cnt[b * 256 + t];
  s[t] = v;
  __syncthreads();
  for (int d = 1; d < 256; d <<= 1) {
    int x = (t >= d) ? s[t - d] : 0;
    __syncthreads();
    s[t] += x;
    __syncthreads();
  }
  nodeOff[b * 256 + t] = chunkOff[b] + s[t] - v;   // exclusive prefix
}

__global__ void fill_scatter(const int* __restrict__ dst,
                             const int* __restrict__ nodeOff,
                             int* __restrict__ fill, int* __restrict__ eid,
                             int ne) {
  int e = blockIdx.x * blockDim.x + threadIdx.x;
  if (e < ne) {
    const int d = dst[e];
    const int p = atomicAdd(&fill[d], 1);
    eid[nodeOff[d] + p] = e;
  }
}

// deg[n] = 1 + sum of incoming edge weights (via CSR, no atomics)
__global__ void deg_csr(const int* __restrict__ nodeOff,
                        const int* __restrict__ cnt,
                        const int* __restrict__ eid,
                        const float* __restrict__ ew,
                        float* __restrict__ deg, int n) {
  int i = blockIdx.x * blockDim.x + threadIdx.x;
  if (i >= n) return;
  float s = 1.0f;
  const int o = nodeOff[i], c = cnt[i];
  for (int j = 0; j < c; ++j) s += ew[eid[o + j]];
  deg[i] = s;
}

__global__ void norm_kernel(const int* __restrict__ src,
                            const int* __restrict__ dst,
                            const float* __restrict__ ew,
                            const float* __restrict__ deg,
                            float* __restrict__ nrm, int ne) {
  int e = blockIdx.x * blockDim.x + threadIdx.x;
  if (e < ne)
    nrm[e] = __frsqrt_rn(deg[src[e]]) * ew[e] * __frsqrt_rn(deg[dst[e]]);
}

// agg[n] = relu( sum_e h[src[e]]*nrm[e] + h[n]/deg[n] + bias );  16 thr/node
__global__ void __launch_bounds__(256)
gather_kernel(const int* __restrict__ nodeOff, const int* __restrict__ cnt,
              const int* __restrict__ eid, const int* __restrict__ src,
              const float* __restrict__ nrm, const float* __restrict__ deg,
              const float* __restrict__ h, const float* __restrict__ bg,
              float* __restrict__ agg) {
  const int t = blockIdx.x * blockDim.x + threadIdx.x;
  const int node = t >> 4;
  const int q = t & 15;
  const float dinv2 = 1.0f / deg[node];
  float4 acc = ((const float4*)(h + (size_t)node * FD))[q];
  acc.x *= dinv2; acc.y *= dinv2; acc.z *= dinv2; acc.w *= dinv2;
  const int o = nodeOff[node], c = cnt[node];
  for (int j = 0; j < c; ++j) {
    const int e = eid[o + j];
    const int s = src[e];
    const float nm = nrm[e];
    __builtin_prefetch(h + (size_t)s * FD, 0, 0);   // global_prefetch_b8
    const float4 hv = ((const float4*)(h + (size_t)s * FD))[q];
    acc.x += hv.x * nm; acc.y += hv.y * nm;
    acc.z += hv.z * nm; acc.w += hv.w * nm;
  }
  const float4 b4 = ((const float4*)bg)[q];
  float4 r;
  r.x = fmaxf(acc.x + b4.x, 0.0f);
  r.y = fmaxf(acc.y + b4.y, 0.0f);
  r.z = fmaxf(acc.z + b4.z, 0.0f);
  r.w = fmaxf(acc.w + b4.w, 0.0f);
  ((float4*)(agg + (size_t)node * FD))[q] = r;
}

// per-graph [sum, mean] pooling (agg already has bias+relu).  One block/graph.
__global__ void __launch_bounds__(256)
pool_kernel(const float* __restrict__ agg, float* __restrict__ pooled) {
  __shared__ float red[256];
  const int g = blockIdx.x;
  const int t = threadIdx.x;
  const int f = t & 63;
  const int sub = t >> 6;
  float sum = 0.0f;
  const long base = (long)g * NPG;
  for (int i = sub * 256; i < sub * 256 + 256; ++i)
    sum += agg[(base + i) * FD + f];
  red[t] = sum;
  __syncthreads();
  if (sub == 0) {
    const float s = red[f] + red[64 + f] + red[128 + f] + red[192 + f];
    pooled[g * 128 + f]      = s;                       // global sum pool
    pooled[g * 128 + 64 + f] = s * (1.0f / (float)NPG); // global avg pool
  }
}

// ---- final MLP head; one block (128 thr) per graph -------------------------
__global__ void __launch_bounds__(128)
head_kernel(const float* __restrict__ pa, const float* __restrict__ pb,
            const float* __restrict__ linker,
            const float* __restrict__ W1, const float* __restrict__ b1,
            const float* __restrict__ W2, const float* __restrict__ b2,
            const float* __restrict__ W3, const float* __restrict__ b3,
            const float* __restrict__ Wcb, const float* __restrict__ bcb,
            const float* __restrict__ Wom, const float* __restrict__ bom,
            const float* __restrict__ Wth, const float* __restrict__ bth,
            const float* __restrict__ Wph, const float* __restrict__ bph,
            float* __restrict__ out) {
  __shared__ float feat[257];
  __shared__ float z1[128];
  __shared__ float z2[64];
  const int g = blockIdx.x;
  const int t = threadIdx.x;
  feat[t]       = pa[g * 128 + t];
  feat[128 + t] = pb[g * 128 + t];
  if (t == 0) feat[256] = linker[g];
  __syncthreads();
  float a1 = b1[t];
  for (int i = 0; i < 257; ++i) a1 += feat[i] * W1[i * 128 + t];
  z1[t] = fmaxf(a1, 0.0f);
  __syncthreads();
  if (t < 64) {
    float a2 = b2[t];
    for (int i = 0; i < 128; ++i) a2 += z1[i] * W2[i * 64 + t];
    z2[t] = fmaxf(a2, 0.0f);
  }
  __syncthreads();
  if (t == 0) {
    float v = b3[0];
    for (int i = 0; i < 64; ++i) v += z2[i] * W3[i];
    out[g] = v;
  } else if (t == 1) {
    float v = bcb[0];
    for (int i = 0; i < 64; ++i) v += z2[i] * Wcb[i];
    out[NGRAPH + g] = v;
  } else if (t < 4) {
    const int j = t - 2;
    float v = bom[j];
    for (int i = 0; i < 64; ++i) v += z2[i] * Wom[i * 2 + j];
    out[2 * NGRAPH + g * 2 + j] = v;
  } else if (t < 6) {
    const int j = t - 4;
    float v = bth[j];
    for (int i = 0; i < 64; ++i) v += z2[i] * Wth[i * 2 + j];
    out[4 * NGRAPH + g * 2 + j] = v;
  } else if (t < 8) {
    const int j = t - 6;
    float v = bph[j];
    for (int i = 0; i < 64; ++i) v += z2[i] * Wph[i * 2 + j];
    out[6 * NGRAPH + g * 2 + j] = v;
  }
}

// ---------------------------------------------------------------------------
extern "C" void kernel_launch(void* const* d_in, const int* in_sizes, int n_in,
                              void* d_out, int out_size, void* d_ws, size_t ws_size,
                              hipStream_t stream) {
  const float* x_a    = (const float*)d_in[0];
  const int*   ei_a   = (const int*)d_in[1];
  const float* ea_a   = (const float*)d_in[2];
  const float* x_b    = (const float*)d_in[4];
  const int*   ei_b   = (const int*)d_in[5];
  const float* ea_b   = (const float*)d_in[6];
  const float* linker = (const float*)d_in[8];

  const float *W1a=(const float*)d_in[9],  *b1a=(const float*)d_in[10];
  const float *W2a=(const float*)d_in[11], *b2a=(const float*)d_in[12];
  const float *Wga=(const float*)d_in[13], *bga=(const float*)d_in[14];
  const float *W1b=(const float*)d_in[15], *b1b=(const float*)d_in[16];
  const float *W2b=(const float*)d_in[17], *b2b=(const float*)d_in[18];
  const float *Wgb=(const float*)d_in[19], *bgb=(const float*)d_in[20];
  const float *Wl1=(const float*)d_in[21], *bl1=(const float*)d_in[22];
  const float *Wl2=(const float*)d_in[23], *bl2=(const float*)d_in[24];
  const float *Wl3=(const float*)d_in[25], *bl3=(const float*)d_in[26];
  const float *Wcb=(const float*)d_in[27], *bcb=(const float*)d_in[28];
  const float *Wom=(const float*)d_in[29], *bom=(const float*)d_in[30];
  const float *Wth=(const float*)d_in[31], *bth=(const float*)d_in[32];
  const float *Wph=(const float*)d_in[33], *bph=(const float*)d_in[34];

  // workspace carve-up
  float* ws   = (float*)d_ws;
  float* h    = ws;                                   // NNODES*64 f
  float* agg  = h    + (size_t)NNODES * FD;           // NNODES*64 f
  float* deg  = agg  + (size_t)NNODES * FD;           // NNODES f
  float* nrm  = deg  + NNODES;                        // NEDGES f
  float* pa   = nrm  + NEDGES;                        // 128*128 f
  float* pb   = pa   + NGRAPH * 128;                  // 128*128 f
  int*   cnt  = (int*)(pb + NGRAPH * 128);            // NNODES i
  int*   fill = cnt  + NNODES;                        // NNODES i
  int*   nOff = fill + NNODES;                        // NNODES i
  int*   eid  = nOff + NNODES;                        // NEDGES i
  int*   cSum = eid  + NEDGES;                        // CHUNKS i
  int*   cOff = cSum + CHUNKS;                        // CHUNKS i

  const float* xs[2]  = {x_a, x_b};
  const int*   eis[2] = {ei_a, ei_b};
  const float* eas[2] = {ea_a, ea_b};
  const float* W1s[2] = {W1a, W1b};  const float* b1s[2] = {b1a, b1b};
  const float* W2s[2] = {W2a, W2b};  const float* b2s[2] = {b2a, b2b};
  const float* Wgs[2] = {Wga, Wgb};  const float* bgs[2] = {bga, bgb};
  float* pooled[2] = {pa, pb};

  for (int br = 0; br < 2; ++br) {
    const int* src = eis[br];
    const int* dst = eis[br] + NEDGES;
    const float* ew = eas[br];

    mlp_branch<<<NNODES / 64, 128, 0, stream>>>(
        xs[br], W1s[br], b1s[br], W2s[br], b2s[br], Wgs[br], h);

    // CSR-by-dst build
    zero_ints<<<(2 * NNODES) / 256, 256, 0, stream>>>(cnt, 2 * NNODES); // cnt+fill
    cnt_scatter<<<NEDGES / 256, 256, 0, stream>>>(dst, cnt, NEDGES);
    chunk_reduce<<<CHUNKS, 256, 0, stream>>>(cnt, cSum);
    chunk_scan<<<1, CHUNKS, 0, stream>>>(cSum, cOff);
    node_scan<<<CHUNKS, 256, 0, stream>>>(cnt, cOff, nOff);
    fill_scatter<<<NEDGES / 256, 256, 0, stream>>>(dst, nOff, fill, eid, NEDGES);

    // degrees, edge norms, gather-aggregate (bias+relu fused)
    deg_csr<<<NNODES / 256, 256, 0, stream>>>(nOff, cnt, eid, ew, deg, NNODES);
    norm_kernel<<<NEDGES / 256, 256, 0, stream>>>(src, dst, ew, deg, nrm, NEDGES);
    gather_kernel<<<(NNODES * 16) / 256, 256, 0, stream>>>(
        nOff, cnt, eid, src, nrm, deg, h, bgs[br], agg);

    pool_kernel<<<NGRAPH, 256, 0, stream>>>(agg, pooled[br]);
  }

  head_kernel<<<NGRAPH, 128, 0, stream>>>(
      pa, pb, linker, Wl1, bl1, Wl2, bl2, Wl3, bl3,
      Wcb, bcb, Wom, bom, Wth, bth, Wph, bph, (float*)d_out);
}